// GAT_67508295958887
// MI455X (gfx1250) — compile-verified
//
#include <hip/hip_runtime.h>
#include <hip/hip_bf16.h>

typedef __attribute__((ext_vector_type(2))) float v2f;
typedef __attribute__((ext_vector_type(8))) float v8f;

#define SLOPE 0.2f

// ---------------------------------------------------------------------------
// WMMA f32 GEMM:  C[M,Nout] = A[M,K] @ B[K,Np]   (B zero-padded to Np cols)
// One wave computes a 16x64 strip of C: 4 accumulators sharing each A
// fragment -> 4 back-to-back v_wmma_f32_16x16x4_f32 per K-step, no
// predication anywhere in the loop (EXEC stays all-1s, no saveexec).
// A-frag (16x4 f32, 2 VGPRs): lanes 0-15 -> M=lane,   K=k+0/k+1;
//                             lanes 16-31 -> M=lane-16, K=k+2/k+3.
// C (8 VGPRs): VGPR r: lanes 0-15 -> (M=r, N=lane); 16-31 -> (M=r+8, N=lane-16).
// Requires M % 16 == 0 (50000 = 3125*16) and Np % 64 == 0.
// ---------------------------------------------------------------------------
__global__ __launch_bounds__(32) void gat_wmma_gemm_f32(
    const float* __restrict__ A, const float* __restrict__ B,
    float* __restrict__ C, int M, int Np, int K, int Nout)
{
    const int lane  = threadIdx.x & 31;
    const int l15   = lane & 15;
    const int khalf = (lane >> 4) * 2;            // 0 or 2
    const int mt    = blockIdx.x;                 // 16-row tile
    const int col0  = blockIdx.y * 64;            // 64-col strip
    const int row   = mt * 16 + l15;              // always < M

    const float* __restrict__ Ap = A + (size_t)row * K + khalf;
    const float* __restrict__ Bp = B + (size_t)khalf * Np + col0 + l15;

    v8f acc[4];
    #pragma unroll
    for (int t = 0; t < 4; ++t)
        acc[t] = (v8f){0.f, 0.f, 0.f, 0.f, 0.f, 0.f, 0.f, 0.f};

    for (int k = 0; k < K; k += 4) {
        v2f a;
        a.x = Ap[0];
        a.y = Ap[1];
        v2f b0, b1, b2, b3;
        b0.x = Bp[0];   b0.y = Bp[Np + 0];
        b1.x = Bp[16];  b1.y = Bp[Np + 16];
        b2.x = Bp[32];  b2.y = Bp[Np + 32];
        b3.x = Bp[48];  b3.y = Bp[Np + 48];
        acc[0] = __builtin_amdgcn_wmma_f32_16x16x4_f32(false, a, false, b0, (short)0, acc[0], false, false);
        acc[1] = __builtin_amdgcn_wmma_f32_16x16x4_f32(false, a, false, b1, (short)0, acc[1], false, false);
        acc[2] = __builtin_amdgcn_wmma_f32_16x16x4_f32(false, a, false, b2, (short)0, acc[2], false, false);
        acc[3] = __builtin_amdgcn_wmma_f32_16x16x4_f32(false, a, false, b3, (short)0, acc[3], false, false);
        Ap += 4;
        Bp += (size_t)4 * Np;
    }

    const int srow0 = mt * 16 + (lane >> 4) * 8;
    #pragma unroll
    for (int t = 0; t < 4; ++t) {
        const int col = col0 + t * 16 + l15;
        if (col < Nout) {                          // only hits for padded layer-1 tail
            #pragma unroll
            for (int r = 0; r < 8; ++r)
                C[(size_t)(srow0 + r) * Nout + col] = acc[t][r];
        }
    }
}

// Zero-pad W[K,N] -> Wp[K,Np]
__global__ __launch_bounds__(256) void gat_pad_weights(
    const float* __restrict__ W, float* __restrict__ Wp, int K, int N, int Np)
{
    int t = blockIdx.x * blockDim.x + threadIdx.x;
    if (t >= K * Np) return;
    const int r = t / Np, c = t - r * Np;
    Wp[t] = (c < N) ? W[r * N + c] : 0.f;
}

// ---------------------------------------------------------------------------
// Per-node attention scalars: el[n,h] = sum_d h[n,h,d]*al[h,d]; same for er.
// One 64-thread block per (node, head); coalesced loads + LDS tree reduce.
// ---------------------------------------------------------------------------
__global__ __launch_bounds__(64) void gat_node_attn(
    const float* __restrict__ h, const float* __restrict__ al,
    const float* __restrict__ ar, float* __restrict__ el,
    float* __restrict__ er, int H, int D)
{
    __shared__ float sl[64];
    __shared__ float sr[64];
    const int n = blockIdx.x, hh = blockIdx.y, d = threadIdx.x;

    float x = 0.f, a = 0.f, b = 0.f;
    if (d < D) {
        x = h[((size_t)n * H + hh) * D + d];
        a = al[hh * D + d];
        b = ar[hh * D + d];
    }
    sl[d] = x * a;
    sr[d] = x * b;
    __syncthreads();
    #pragma unroll
    for (int off = 32; off > 0; off >>= 1) {
        if (d < off) { sl[d] += sl[d + off]; sr[d] += sr[d + off]; }
        __syncthreads();
    }
    if (d == 0) {
        el[(size_t)n * H + hh] = sl[0];
        er[(size_t)n * H + hh] = sr[0];
    }
}

// Monotone float <-> ordered-uint mapping (atomicMax correct for any sign).
__device__ __forceinline__ unsigned f2ord(float f) {
    unsigned u = __float_as_uint(f);
    return (u & 0x80000000u) ? ~u : (u | 0x80000000u);
}
__device__ __forceinline__ float ord2f(unsigned u) {
    unsigned b = (u & 0x80000000u) ? (u ^ 0x80000000u) : ~u;
    return __uint_as_float(b);
}

// Phase 1: e = leakyrelu(el[src]+er[dst]); segment-max into mU[dst].
__global__ __launch_bounds__(256) void gat_edge_max(
    const int* __restrict__ src, const int* __restrict__ dst,
    const float* __restrict__ el, const float* __restrict__ er,
    float* __restrict__ eT, unsigned* __restrict__ mU, int E, int H)
{
    int t = blockIdx.x * blockDim.x + threadIdx.x;
    if (t >= E * H) return;
    const int e = t / H, hh = t - e * H;
    float v = el[(size_t)src[e] * H + hh] + er[(size_t)dst[e] * H + hh];
    v = (v > 0.f) ? v : SLOPE * v;
    eT[t] = v;
    atomicMax(&mU[(size_t)dst[e] * H + hh], f2ord(v));
}

// Phase 2: alpha = exp(e - m[dst]); segment-sum into s[dst].
__global__ __launch_bounds__(256) void gat_edge_exp(
    const int* __restrict__ dst, const float* __restrict__ eT,
    const unsigned* __restrict__ mU, float* __restrict__ alphaT,
    float* __restrict__ s, int E, int H)
{
    int t = blockIdx.x * blockDim.x + threadIdx.x;
    if (t >= E * H) return;
    const int e = t / H, hh = t - e * H;
    const float m = ord2f(mU[(size_t)dst[e] * H + hh]);
    const float a = __expf(eT[t] - m);
    alphaT[t] = a;
    atomicAdd(&s[(size_t)dst[e] * H + hh], a);
}

// Phase 3: out[dst] += (alpha/s[dst]) * h[src].  One wave per edge; lanes
// stride the H*D payload with global_atomic_add_f32 (no return).
// Head index is v>>6: exact for layer0 (D=64) and layer1 (H=1, HD=40<64).
__global__ __launch_bounds__(256) void gat_edge_aggregate(
    const int* __restrict__ src, const int* __restrict__ dst,
    const float* __restrict__ h, const float* __restrict__ alphaT,
    const float* __restrict__ s, float* __restrict__ out,
    int E, int H, int HD)
{
    const int wid  = (blockIdx.x * blockDim.x + threadIdx.x) >> 5;
    const int lane = threadIdx.x & 31;
    if (wid >= E) return;
    const int sN = src[wid], dN = dst[wid];
    const float* __restrict__ hs = h + (size_t)sN * HD;
    float* __restrict__ od = out + (size_t)dN * HD;
    for (int v = lane; v < HD; v += 32) {
        const int hh = v >> 6;
        const float a = alphaT[(size_t)wid * H + hh] / s[(size_t)dN * H + hh];
        atomicAdd(&od[v], a * hs[v]);
    }
}

// ---------------------------------------------------------------------------
extern "C" void kernel_launch(void* const* d_in, const int* in_sizes, int n_in,
                              void* d_out, int out_size, void* d_ws, size_t ws_size,
                              hipStream_t stream)
{
    const float* feat = (const float*)d_in[0];
    const int*   src  = (const int*)  d_in[1];
    const int*   dst  = (const int*)  d_in[2];
    const float* W0   = (const float*)d_in[3];
    const float* al0  = (const float*)d_in[4];
    const float* ar0  = (const float*)d_in[5];
    const float* W1   = (const float*)d_in[6];
    const float* al1  = (const float*)d_in[7];
    const float* ar1  = (const float*)d_in[8];
    float* out = (float*)d_out;

    const int FIN = 128, H0 = 4, D0 = 64, HD0 = H0 * D0, NC = 40, NCP = 64;
    const int N = in_sizes[0] / FIN;   // 50000 (multiple of 16)
    const int E = in_sizes[1];         // 800000

    // ---- workspace layout (256B aligned slices) ----
    char* p = (char*)d_ws;
    auto alloc = [&](size_t bytes) -> void* {
        void* r = (void*)p;
        p += (bytes + 255) & ~(size_t)255;
        return r;
    };
    float*    h0      = (float*)   alloc((size_t)N * HD0 * 4);
    float*    agg0    = (float*)   alloc((size_t)N * HD0 * 4);
    float*    h1      = (float*)   alloc((size_t)N * NC  * 4);
    float*    W1p     = (float*)   alloc((size_t)HD0 * NCP * 4);  // padded W1
    float*    el0     = (float*)   alloc((size_t)N * H0 * 4);
    float*    er0     = (float*)   alloc((size_t)N * H0 * 4);
    unsigned* mU0     = (unsigned*)alloc((size_t)N * H0 * 4);
    float*    s0      = (float*)   alloc((size_t)N * H0 * 4);
    float*    eT0     = (float*)   alloc((size_t)E * H0 * 4);
    float*    alphaT0 = (float*)   alloc((size_t)E * H0 * 4);
    float*    el1     = (float*)   alloc((size_t)N * 4);
    float*    er1     = (float*)   alloc((size_t)N * 4);
    unsigned* mU1     = (unsigned*)alloc((size_t)N * 4);
    float*    s1      = (float*)   alloc((size_t)N * 4);
    float*    eT1     = (float*)   alloc((size_t)E * 4);
    float*    alphaT1 = (float*)   alloc((size_t)E * 4);

    // ================= Layer 0 =================
    // h0[N,256] = feat[N,128] @ W0[128,256]   (Np = Nout = 256, no padding)
    {
        dim3 grid(N / 16, HD0 / 64);
        gat_wmma_gemm_f32<<<grid, 32, 0, stream>>>(feat, W0, h0, N, HD0, FIN, HD0);
    }
    {
        dim3 grid(N, H0);
        gat_node_attn<<<grid, 64, 0, stream>>>(h0, al0, ar0, el0, er0, H0, D0);
    }
    hipMemsetAsync(mU0, 0, (size_t)N * H0 * 4, stream);   // ordered-uint -inf
    hipMemsetAsync(s0,  0, (size_t)N * H0 * 4, stream);
    {
        int threads = E * H0;
        gat_edge_max<<<(threads + 255) / 256, 256, 0, stream>>>(
            src, dst, el0, er0, eT0, mU0, E, H0);
        gat_edge_exp<<<(threads + 255) / 256, 256, 0, stream>>>(
            dst, eT0, mU0, alphaT0, s0, E, H0);
    }
    hipMemsetAsync(agg0, 0, (size_t)N * HD0 * 4, stream);
    {
        long long threads = (long long)E * 32;             // one wave per edge
        gat_edge_aggregate<<<(int)((threads + 255) / 256), 256, 0, stream>>>(
            src, dst, h0, alphaT0, s0, agg0, E, H0, HD0);
    }

    // ================= Layer 1 =================
    // Pad W1[256,40] -> W1p[256,64], then h1[N,40] = agg0[N,256] @ W1p.
    {
        int tot = HD0 * NCP;
        gat_pad_weights<<<(tot + 255) / 256, 256, 0, stream>>>(W1, W1p, HD0, NC, NCP);
        dim3 grid(N / 16, NCP / 64);
        gat_wmma_gemm_f32<<<grid, 32, 0, stream>>>(agg0, W1p, h1, N, NCP, HD0, NC);
    }
    {
        dim3 grid(N, 1);
        gat_node_attn<<<grid, 64, 0, stream>>>(h1, al1, ar1, el1, er1, 1, NC);
    }
    hipMemsetAsync(mU1, 0, (size_t)N * 4, stream);
    hipMemsetAsync(s1,  0, (size_t)N * 4, stream);
    {
        gat_edge_max<<<(E + 255) / 256, 256, 0, stream>>>(
            src, dst, el1, er1, eT1, mU1, E, 1);
        gat_edge_exp<<<(E + 255) / 256, 256, 0, stream>>>(
            dst, eT1, mU1, alphaT1, s1, E, 1);
    }
    // H1==1 -> mean over heads is identity: accumulate directly into d_out.
    hipMemsetAsync(out, 0, (size_t)N * NC * 4, stream);
    {
        long long threads = (long long)E * 32;
        gat_edge_aggregate<<<(int)((threads + 255) / 256), 256, 0, stream>>>(
            src, dst, h1, alphaT1, s1, out, E, 1, NC);
    }
}